// Loss_30812095381806
// MI455X (gfx1250) — compile-verified
//
#include <hip/hip_runtime.h>

typedef float v2f __attribute__((ext_vector_type(2)));
typedef float v8f __attribute__((ext_vector_type(8)));

#define B_SZ 8192
#define T_SZ 512
#define C_SZ 7

#define LOG2E 1.44269504088896340736f
#define LN2   0.69314718055994530942f

// raw hardware ops: v_exp_f32 (2^x), v_log_f32 (log2 x), v_rcp_f32
__device__ __forceinline__ float fexp(float x)  { return __builtin_amdgcn_exp2f(x * LOG2E); }
__device__ __forceinline__ float flog2(float x) { return __builtin_amdgcn_logf(x); }
__device__ __forceinline__ float frcp(float x)  { return __builtin_amdgcn_rcpf(x); }

// xor-16 lane swap within wave32 (SWAPX16): and=0x1f, or=0, xor=0x10 -> 0x401F
__device__ __forceinline__ float swz16(float x) {
    int r = __builtin_amdgcn_ds_swizzle(__float_as_int(x), 0x401F);
    return __int_as_float(r);
}

// ---------------------------------------------------------------------------
// Kernel 1: numerator (labeled path score) per batch element.
// ---------------------------------------------------------------------------
__global__ __launch_bounds__(256) void crf_numerator(
    const float* __restrict__ em, const float* __restrict__ trans,
    const float* __restrict__ startT, const float* __restrict__ endT,
    const int* __restrict__ mask, const int* __restrict__ labels,
    float* __restrict__ num) {
    __shared__ float sTr[C_SZ * C_SZ];
    __shared__ float sStart[C_SZ];
    __shared__ float sEnd[C_SZ];
    int tid = threadIdx.x;
    if (tid < C_SZ * C_SZ) sTr[tid] = trans[tid];
    if (tid < C_SZ) { sStart[tid] = startT[tid]; sEnd[tid] = endT[tid]; }
    __syncthreads();

    int b = blockIdx.x * blockDim.x + tid;
    if (b >= B_SZ) return;

    const int*   lb = labels + (size_t)b * T_SZ;
    const int*   mk = mask   + (size_t)b * T_SZ;
    const float* eb = em     + (size_t)b * T_SZ * C_SZ;

    int last = lb[0];
    float score = sStart[last] + eb[last];
    for (int t = 1; t < T_SZ; ++t) {
        int l = lb[t];
        int m = mk[t];
        float add = sTr[last * C_SZ + l] + eb[t * C_SZ + l];
        if (m > 0) { score += add; last = l; }
    }
    score += sEnd[last];
    num[b] = score;
}

// ---------------------------------------------------------------------------
// Kernel 2: forward algorithm in probability domain via f32 WMMA.
// One wave handles 16 batch rows. D-layout (lane = batch col, VGPR = state)
// feeds the next step's B operand with just four xor-16 swizzles.
// Running scale accumulated in log2 domain; converted to ln once at the end.
// ---------------------------------------------------------------------------
__global__ __launch_bounds__(256) void crf_forward_wmma(
    const float* __restrict__ em, const float* __restrict__ trans,
    const float* __restrict__ startT, const float* __restrict__ endT,
    const int* __restrict__ mask, const float* __restrict__ num,
    float* __restrict__ outdiff) {
    int lane = threadIdx.x & 31;
    int waveInBlock = threadIdx.x >> 5;
    int wave = blockIdx.x * 8 + waveInBlock;
    int half = lane >> 4;
    int n    = lane & 15;
    int row  = wave * 16 + n;   // batch row for this lane's column

    // A = P^T, M (rows) = dest state j (0..6 valid), K = source state i.
    // A 16x4 layout: lanes 0-15 hold K={0,1}, lanes 16-31 hold K={2,3}.
    int j = n;
    v2f a0, a1;
    {
        float v00 = 0.f, v01 = 0.f, v10 = 0.f, v11 = 0.f;
        int i0 = half ? 2 : 0;   // a0 covers K = i0, i0+1
        int i1 = half ? 6 : 4;   // a1 covers K = i1, i1+1 (i=7 -> zero pad)
        if (j < C_SZ) {
            v00 = fexp(trans[i0 * C_SZ + j]);
            v01 = fexp(trans[(i0 + 1) * C_SZ + j]);
            v10 = fexp(trans[i1 * C_SZ + j]);
            if (i1 + 1 < C_SZ) v11 = fexp(trans[(i1 + 1) * C_SZ + j]);
        }
        a0[0] = v00; a0[1] = v01;
        a1[0] = v10; a1[1] = v11;
    }

    const float* erow = em   + (size_t)row * T_SZ * C_SZ;
    const int*   mrow = mask + (size_t)row * T_SZ;

    // init: alpha0 = exp(start + e0), normalized; acc2 = running log2 scale
    v8f d;
    float acc2;
    {
        float u[C_SZ];
#pragma unroll
        for (int k = 0; k < C_SZ; ++k) u[k] = fexp(startT[k] + erow[k]);
        float s = ((u[0] + u[1]) + (u[2] + u[3])) + ((u[4] + u[5]) + u[6]);
        float inv = frcp(s);
#pragma unroll
        for (int k = 0; k < C_SZ; ++k) d[k] = u[k] * inv;
        d[7] = 0.f;
        acc2 = flog2(s);
    }

    for (int t = 1; t < T_SZ; ++t) {
        // Build B operands (K x 16) from current alpha (lanes 0-15 own data;
        // lanes 16-31 take K=2,3 / 6,7 rows swizzled from partner lane).
        float t2 = swz16(d[2]);
        float t3 = swz16(d[3]);
        float t6 = swz16(d[6]);
        float t7 = swz16(d[7]);
        v2f b0, b1;
        b0[0] = half ? t2 : d[0];
        b0[1] = half ? t3 : d[1];
        b1[0] = half ? t6 : d[4];
        b1[1] = half ? t7 : d[5];

        v8f c = {0.f, 0.f, 0.f, 0.f, 0.f, 0.f, 0.f, 0.f};
        c = __builtin_amdgcn_wmma_f32_16x16x4_f32(false, a0, false, b0,
                                                  (short)0, c, false, false);
        c = __builtin_amdgcn_wmma_f32_16x16x4_f32(false, a1, false, b1,
                                                  (short)0, c, false, false);

        // emission multiply + renormalize (pure per-lane: lane holds all states)
        const float* ep = erow + t * C_SZ;
        float nd[C_SZ];
#pragma unroll
        for (int k = 0; k < C_SZ; ++k) nd[k] = c[k] * fexp(ep[k]);
        float s = ((nd[0] + nd[1]) + (nd[2] + nd[3])) + ((nd[4] + nd[5]) + nd[6]);
        float inv   = frcp(s);
        float nacc2 = acc2 + flog2(s);
        int mt = mrow[t];
        bool upd = mt > 0;
#pragma unroll
        for (int k = 0; k < C_SZ; ++k) d[k] = upd ? nd[k] * inv : d[k];
        acc2 = upd ? nacc2 : acc2;
        // d[7] stays 0 (never written)
    }

    if (lane < 16) {
        float s = 0.f;
#pragma unroll
        for (int k = 0; k < C_SZ; ++k) s += d[k] * fexp(endT[k]);
        float denom = (acc2 + flog2(s)) * LN2;
        outdiff[row] = denom - num[row];
    }
}

// ---------------------------------------------------------------------------
// Kernel 3: deterministic mean reduction over B values.
// ---------------------------------------------------------------------------
__global__ __launch_bounds__(256) void crf_reduce_mean(
    const float* __restrict__ diff, float* __restrict__ out) {
    __shared__ float sm[256];
    float s = 0.f;
    for (int i = threadIdx.x; i < B_SZ; i += 256) s += diff[i];
    sm[threadIdx.x] = s;
    __syncthreads();
    for (int off = 128; off > 0; off >>= 1) {
        if (threadIdx.x < off) sm[threadIdx.x] += sm[threadIdx.x + off];
        __syncthreads();
    }
    if (threadIdx.x == 0) out[0] = sm[0] * (1.0f / (float)B_SZ);
}

extern "C" void kernel_launch(void* const* d_in, const int* in_sizes, int n_in,
                              void* d_out, int out_size, void* d_ws, size_t ws_size,
                              hipStream_t stream) {
    const float* em = (const float*)d_in[0];
    const float* tr = (const float*)d_in[1];
    const float* st = (const float*)d_in[2];
    const float* en = (const float*)d_in[3];
    const int*   mk = (const int*)d_in[4];
    const int*   lb = (const int*)d_in[5];
    float* out  = (float*)d_out;
    float* num  = (float*)d_ws;          // B floats: path scores
    float* diff = num + B_SZ;            // B floats: denom - num

    crf_numerator<<<B_SZ / 256, 256, 0, stream>>>(em, tr, st, en, mk, lb, num);
    crf_forward_wmma<<<B_SZ / 128, 256, 0, stream>>>(em, tr, st, en, mk, num, diff);
    crf_reduce_mean<<<1, 256, 0, stream>>>(diff, out);
}